// LIFAct_17051020165839
// MI455X (gfx1250) — compile-verified
//
#include <hip/hip_runtime.h>

// LIF neuron forward scan, STEP=4, on [4, 64, 128, 32, 32] fp32.
// Pure streaming: 134 MB in + 134 MB out, zero reuse -> HBM-bandwidth bound
// (~11.5 us floor at 23.3 TB/s). Strategy: b128 non-temporal loads/stores,
// 4 outstanding loads per thread (one per time step), register-resident
// 4-step scan, NT hints to bypass L2 pollution.

typedef float v4f __attribute__((ext_vector_type(4)));

#define LIF_DECAY  0.25f
#define LIF_THRESH 0.5f   // (mem / V_TH) > 0.5 with V_TH = 1.0
#define LIF_STEPS  4

__global__ __launch_bounds__(256) void lif_fwd_kernel(
    const v4f* __restrict__ x,   // [STEP * n4] float4, time-major
    v4f* __restrict__ out,       // [STEP * n4] float4
    int n4)                      // float4 sites per time step
{
    const int i = blockIdx.x * blockDim.x + threadIdx.x;
    if (i >= n4) return;

    // Issue all 4 time-step loads up front: 4 outstanding global_load_b128
    // per lane. NT hint: data is touched exactly once.
    v4f xt[LIF_STEPS];
#pragma unroll
    for (int t = 0; t < LIF_STEPS; ++t) {
        xt[t] = __builtin_nontemporal_load(x + (size_t)t * n4 + i);
    }

    // Register-resident LIF scan over the 4 time steps.
    v4f mem = {0.0f, 0.0f, 0.0f, 0.0f};
#pragma unroll
    for (int t = 0; t < LIF_STEPS; ++t) {
        v4f spike;
#pragma unroll
        for (int c = 0; c < 4; ++c) {
            float m = mem[c] * LIF_DECAY + xt[t][c];   // leaky integrate
            const bool fired = m > LIF_THRESH;          // hard threshold (fwd)
            spike[c] = fired ? 1.0f : 0.0f;             // v_cndmask off v_cmp
            mem[c]   = fired ? 0.0f : m;                // reset where spiked
        }
        __builtin_nontemporal_store(spike, out + (size_t)t * n4 + i);
    }
}

extern "C" void kernel_launch(void* const* d_in, const int* in_sizes, int n_in,
                              void* d_out, int out_size, void* d_ws, size_t ws_size,
                              hipStream_t stream) {
    (void)n_in; (void)d_ws; (void)ws_size; (void)out_size;

    const v4f* x  = (const v4f*)d_in[0];
    v4f* out      = (v4f*)d_out;

    // in_sizes[0] = STEP * B * C * H * W = 4 * 64 * 128 * 32 * 32 = 33,554,432
    const int total    = in_sizes[0];
    const int per_step = total / LIF_STEPS;   // elements per time slice
    const int n4       = per_step / 4;        // float4 sites per time slice
                                              // (W=32 -> per-row float4 count is exact)

    const int threads = 256;                  // 8 wave32 waves per block
    const int blocks  = (n4 + threads - 1) / threads;   // 8192 blocks

    lif_fwd_kernel<<<blocks, threads, 0, stream>>>(x, out, n4);
}